// UNsamplingMoudle_51144470561308
// MI455X (gfx1250) — compile-verified
//
#include <hip/hip_runtime.h>
#include <hip/hip_bf16.h>

typedef float v2f __attribute__((ext_vector_type(2)));
typedef float v8f __attribute__((ext_vector_type(8)));

#define B_   8
#define N_   8192
#define S_   2048
#define C1_  128
#define C2_  256
#define CIN_ 384
#define H_   256
#define O_   128
#define M_   (B_ * N_)

// ---------------------------------------------------------------------------
// 1) 3-nearest-neighbor search: per query point, scan all S=2048 source points
//    (tiled through LDS), keep 3 smallest squared distances + indices, emit
//    normalized inverse-distance weights.
// ---------------------------------------------------------------------------
__global__ __launch_bounds__(128) void knn3_kernel(const float* __restrict__ x1,
                                                   const float* __restrict__ x2,
                                                   int* __restrict__ idx,
                                                   float* __restrict__ wgt) {
  __shared__ float sx[512 * 3];
  const int tid = threadIdx.x;
  const int blocksPerBatch = N_ / 128;
  const int b = blockIdx.x / blocksPerBatch;
  const int n = (blockIdx.x % blocksPerBatch) * 128 + tid;
  const int m = b * N_ + n;
  const float qx = x1[m * 3 + 0];
  const float qy = x1[m * 3 + 1];
  const float qz = x1[m * 3 + 2];
  const float* x2b = x2 + (size_t)b * S_ * 3;

  float d0 = 3.0e38f, d1 = 3.0e38f, d2 = 3.0e38f;
  int i0 = 0, i1 = 0, i2 = 0;

  for (int t = 0; t < S_ / 512; ++t) {
    for (int i = tid; i < 512 * 3; i += 128) sx[i] = x2b[t * 512 * 3 + i];
    __syncthreads();
    for (int j = 0; j < 512; ++j) {
      const float dx = qx - sx[j * 3 + 0];
      const float dy = qy - sx[j * 3 + 1];
      const float dz = qz - sx[j * 3 + 2];
      const float d = dx * dx + dy * dy + dz * dz;
      const int s = t * 512 + j;
      if (d < d2) {
        if (d < d1) {
          d2 = d1; i2 = i1;
          if (d < d0) { d1 = d0; i1 = i0; d0 = d; i0 = s; }
          else        { d1 = d;  i1 = s; }
        } else { d2 = d; i2 = s; }
      }
    }
    __syncthreads();
  }

  const float r0 = 1.0f / (d0 + 1e-8f);
  const float r1 = 1.0f / (d1 + 1e-8f);
  const float r2 = 1.0f / (d2 + 1e-8f);
  const float inv = 1.0f / (r0 + r1 + r2);
  idx[m * 3 + 0] = i0; idx[m * 3 + 1] = i1; idx[m * 3 + 2] = i2;
  wgt[m * 3 + 0] = r0 * inv; wgt[m * 3 + 1] = r1 * inv; wgt[m * 3 + 2] = r2 * inv;
}

// ---------------------------------------------------------------------------
// 2) Interpolation: interp[m][c] = sum_j w_j * feature2[b][idx_j][c]
//    One block per point, channels across threads (fully coalesced).
// ---------------------------------------------------------------------------
__global__ __launch_bounds__(C2_) void interp_kernel(const float* __restrict__ f2,
                                                     const int* __restrict__ idx,
                                                     const float* __restrict__ wgt,
                                                     float* __restrict__ interp) {
  const int m = blockIdx.x;
  const int b = m / N_;
  const int c = threadIdx.x;
  const int i0 = idx[m * 3 + 0], i1 = idx[m * 3 + 1], i2 = idx[m * 3 + 2];
  const float w0 = wgt[m * 3 + 0], w1v = wgt[m * 3 + 1], w2v = wgt[m * 3 + 2];
  const float* base = f2 + (size_t)b * S_ * C2_;
  interp[(size_t)m * C2_ + c] = w0  * base[(size_t)i0 * C2_ + c]
                              + w1v * base[(size_t)i1 * C2_ + c]
                              + w2v * base[(size_t)i2 * C2_ + c];
}

// ---------------------------------------------------------------------------
// Fragment conventions for V_WMMA_F32_16X16X4_F32 (wave32):
//   A (16x4, MxK):  lane holds row m = lane&15; VGPR pair = K { kb, kb+1 },
//                   kb = 2*(lane>>4)            -> one float2 load.
//   B (4x16, KxN):  mirrored; lane holds col n = lane&15, same K pairing.
//   C/D (16x16):    acc[r] is row (r + 8*(lane>>4)), col lane&15.
// ---------------------------------------------------------------------------

// 3) GEMM1: Y1[m][n] = sum_k Xcat[m][k] * w1[n][k] + b1[n]
//    Xcat = [feature1 | interp] read from two sources (no concat copy).
//    8 waves/block: 2 (M) x 4 (N); wave tile 16x64 (4 accumulators).
__global__ __launch_bounds__(256) void gemm1_wmma(const float* __restrict__ f1,
                                                  const float* __restrict__ interp,
                                                  const float* __restrict__ w1,
                                                  const float* __restrict__ b1,
                                                  float* __restrict__ Y1) {
  const int tid  = threadIdx.x;
  const int wave = tid >> 5;
  const int lane = tid & 31;
  const int wm = wave >> 2;       // 0..1
  const int wn = wave & 3;        // 0..3
  const int lr = lane & 15;
  const int hi = lane >> 4;
  const int mBase = blockIdx.x * 32 + wm * 16;
  const int rowA  = mBase + lr;
  const int nBase = wn * 64;

  v8f acc[4] = {v8f{}, v8f{}, v8f{}, v8f{}};

  const float* aP0 = f1     + (size_t)rowA * C1_ + hi * 2;
  const float* aP1 = interp + (size_t)rowA * C2_ + hi * 2;
  const float* bR0 = w1 + (size_t)(nBase +  0 + lr) * CIN_ + hi * 2;
  const float* bR1 = w1 + (size_t)(nBase + 16 + lr) * CIN_ + hi * 2;
  const float* bR2 = w1 + (size_t)(nBase + 32 + lr) * CIN_ + hi * 2;
  const float* bR3 = w1 + (size_t)(nBase + 48 + lr) * CIN_ + hi * 2;

#pragma unroll 8
  for (int k = 0; k < C1_; k += 4) {   // feature1 half of K
    const v2f a  = *(const v2f*)(aP0 + k);
    const v2f b0 = *(const v2f*)(bR0 + k);
    const v2f b1f = *(const v2f*)(bR1 + k);
    const v2f b2f = *(const v2f*)(bR2 + k);
    const v2f b3f = *(const v2f*)(bR3 + k);
    acc[0] = __builtin_amdgcn_wmma_f32_16x16x4_f32(false, a, false, b0,  0, acc[0], false, false);
    acc[1] = __builtin_amdgcn_wmma_f32_16x16x4_f32(false, a, false, b1f, 0, acc[1], false, false);
    acc[2] = __builtin_amdgcn_wmma_f32_16x16x4_f32(false, a, false, b2f, 0, acc[2], false, false);
    acc[3] = __builtin_amdgcn_wmma_f32_16x16x4_f32(false, a, false, b3f, 0, acc[3], false, false);
  }
#pragma unroll 8
  for (int k = 0; k < C2_; k += 4) {   // interpolated half of K
    const v2f a  = *(const v2f*)(aP1 + k);
    const v2f b0 = *(const v2f*)(bR0 + C1_ + k);
    const v2f b1f = *(const v2f*)(bR1 + C1_ + k);
    const v2f b2f = *(const v2f*)(bR2 + C1_ + k);
    const v2f b3f = *(const v2f*)(bR3 + C1_ + k);
    acc[0] = __builtin_amdgcn_wmma_f32_16x16x4_f32(false, a, false, b0,  0, acc[0], false, false);
    acc[1] = __builtin_amdgcn_wmma_f32_16x16x4_f32(false, a, false, b1f, 0, acc[1], false, false);
    acc[2] = __builtin_amdgcn_wmma_f32_16x16x4_f32(false, a, false, b2f, 0, acc[2], false, false);
    acc[3] = __builtin_amdgcn_wmma_f32_16x16x4_f32(false, a, false, b3f, 0, acc[3], false, false);
  }

#pragma unroll
  for (int t = 0; t < 4; ++t) {
    const int n = nBase + t * 16 + lr;
    const float bias = b1[n];
#pragma unroll
    for (int r = 0; r < 8; ++r) {
      const int mo = mBase + r + 8 * hi;
      Y1[(size_t)mo * H_ + n] = acc[t][r] + bias;
    }
  }
}

// 5) GEMM2: Y2[m][n] = sum_k relu(scale1[k]*Y1[m][k]+shift1[k]) * w2[n][k] + b2[n]
//    BN+ReLU fused into the A-fragment load. Wave tile 16x32 (2 accumulators).
__global__ __launch_bounds__(256) void gemm2_wmma(const float* __restrict__ Y1,
                                                  const float* __restrict__ sc1,
                                                  const float* __restrict__ sh1,
                                                  const float* __restrict__ w2,
                                                  const float* __restrict__ b2,
                                                  float* __restrict__ Y2) {
  const int tid  = threadIdx.x;
  const int wave = tid >> 5;
  const int lane = tid & 31;
  const int wm = wave >> 2;       // 0..1
  const int wn = wave & 3;        // 0..3
  const int lr = lane & 15;
  const int hi = lane >> 4;
  const int mBase = blockIdx.x * 32 + wm * 16;
  const int rowA  = mBase + lr;
  const int nBase = wn * 32;

  v8f acc[2] = {v8f{}, v8f{}};

  const float* aP  = Y1 + (size_t)rowA * H_ + hi * 2;
  const float* scP = sc1 + hi * 2;
  const float* shP = sh1 + hi * 2;
  const float* bR0 = w2 + (size_t)(nBase +  0 + lr) * H_ + hi * 2;
  const float* bR1 = w2 + (size_t)(nBase + 16 + lr) * H_ + hi * 2;

#pragma unroll 8
  for (int k = 0; k < H_; k += 4) {
    const v2f y = *(const v2f*)(aP + k);
    const v2f s = *(const v2f*)(scP + k);
    const v2f h = *(const v2f*)(shP + k);
    v2f a;
    a.x = fmaxf(0.0f, y.x * s.x + h.x);
    a.y = fmaxf(0.0f, y.y * s.y + h.y);
    const v2f b0 = *(const v2f*)(bR0 + k);
    const v2f b1f = *(const v2f*)(bR1 + k);
    acc[0] = __builtin_amdgcn_wmma_f32_16x16x4_f32(false, a, false, b0,  0, acc[0], false, false);
    acc[1] = __builtin_amdgcn_wmma_f32_16x16x4_f32(false, a, false, b1f, 0, acc[1], false, false);
  }

#pragma unroll
  for (int t = 0; t < 2; ++t) {
    const int n = nBase + t * 16 + lr;
    const float bias = b2[n];
#pragma unroll
    for (int r = 0; r < 8; ++r) {
      const int mo = mBase + r + 8 * hi;
      Y2[(size_t)mo * O_ + n] = acc[t][r] + bias;
    }
  }
}

// ---------------------------------------------------------------------------
// Batch-norm statistics: sum and sum-of-squares per channel, then fold
// gamma/beta into per-channel scale/shift.
// ---------------------------------------------------------------------------
template <int C>
__global__ void stats_kernel(const float* __restrict__ Y, float* __restrict__ sums,
                             int rowsPerBlock) {
  const int c = threadIdx.x;
  const int r0 = blockIdx.x * rowsPerBlock;
  float s = 0.0f, q = 0.0f;
  for (int r = 0; r < rowsPerBlock; ++r) {
    const float v = Y[(size_t)(r0 + r) * C + c];
    s += v;
    q += v * v;
  }
  atomicAdd(&sums[c], s);
  atomicAdd(&sums[C + c], q);
}

__global__ void finalize_kernel(const float* __restrict__ sums, const float* __restrict__ g,
                                const float* __restrict__ be, float* __restrict__ scale,
                                float* __restrict__ shift, int C) {
  const int c = threadIdx.x + blockIdx.x * blockDim.x;
  if (c < C) {
    const float invM = 1.0f / (float)M_;
    const float mean = sums[c] * invM;
    const float var  = sums[C + c] * invM - mean * mean;
    const float sc   = g[c] * rsqrtf(var + 1e-5f);
    scale[c] = sc;
    shift[c] = be[c] - mean * sc;
  }
}

__global__ void zero_kernel(float* __restrict__ p, int n) {
  const int i = threadIdx.x + blockIdx.x * blockDim.x;
  if (i < n) p[i] = 0.0f;
}

// 7) Final BN + ReLU apply -> d_out
__global__ __launch_bounds__(256) void apply_bn_kernel(const float* __restrict__ Y2,
                                                       const float* __restrict__ sc,
                                                       const float* __restrict__ sh,
                                                       float* __restrict__ out) {
  const size_t i = (size_t)blockIdx.x * blockDim.x + threadIdx.x;
  const int c = (int)(i & (O_ - 1));
  out[i] = fmaxf(0.0f, Y2[i] * sc[c] + sh[c]);
}

// ---------------------------------------------------------------------------
extern "C" void kernel_launch(void* const* d_in, const int* in_sizes, int n_in,
                              void* d_out, int out_size, void* d_ws, size_t ws_size,
                              hipStream_t stream) {
  const float* x1  = (const float*)d_in[0];
  const float* x2  = (const float*)d_in[1];
  const float* f1  = (const float*)d_in[2];
  const float* f2  = (const float*)d_in[3];
  const float* w1  = (const float*)d_in[4];
  const float* b1  = (const float*)d_in[5];
  const float* g1  = (const float*)d_in[6];
  const float* be1 = (const float*)d_in[7];
  const float* w2  = (const float*)d_in[8];
  const float* b2  = (const float*)d_in[9];
  const float* g2  = (const float*)d_in[10];
  const float* be2 = (const float*)d_in[11];
  float* out = (float*)d_out;

  char* ws = (char*)d_ws;
  size_t off = 0;
  auto alloc = [&](size_t bytes) -> char* {
    char* p = ws + off;
    off = (off + bytes + 255) & ~(size_t)255;
    return p;
  };
  int*   idx    = (int*)  alloc((size_t)M_ * 3 * sizeof(int));
  float* wgt    = (float*)alloc((size_t)M_ * 3 * sizeof(float));
  float* interp = (float*)alloc((size_t)M_ * C2_ * sizeof(float));
  float* Y1     = (float*)alloc((size_t)M_ * H_ * sizeof(float));
  float* Y2     = (float*)alloc((size_t)M_ * O_ * sizeof(float));
  float* sums1  = (float*)alloc(2 * H_ * sizeof(float));
  float* scale1 = (float*)alloc(H_ * sizeof(float));
  float* shift1 = (float*)alloc(H_ * sizeof(float));
  float* sums2  = (float*)alloc(2 * O_ * sizeof(float));
  float* scale2 = (float*)alloc(O_ * sizeof(float));
  float* shift2 = (float*)alloc(O_ * sizeof(float));

  zero_kernel<<<2, 256, 0, stream>>>(sums1, 2 * H_);
  zero_kernel<<<1, 256, 0, stream>>>(sums2, 2 * O_);

  knn3_kernel<<<B_ * (N_ / 128), 128, 0, stream>>>(x1, x2, idx, wgt);
  interp_kernel<<<M_, C2_, 0, stream>>>(f2, idx, wgt, interp);

  gemm1_wmma<<<M_ / 32, 256, 0, stream>>>(f1, interp, w1, b1, Y1);
  stats_kernel<H_><<<256, H_, 0, stream>>>(Y1, sums1, M_ / 256);
  finalize_kernel<<<1, H_, 0, stream>>>(sums1, g1, be1, scale1, shift1, H_);

  gemm2_wmma<<<M_ / 32, 256, 0, stream>>>(Y1, scale1, shift1, w2, b2, Y2);
  stats_kernel<O_><<<256, O_, 0, stream>>>(Y2, sums2, M_ / 256);
  finalize_kernel<<<1, O_, 0, stream>>>(sums2, g2, be2, scale2, shift2, O_);

  apply_bn_kernel<<<(M_ * O_) / 256, 256, 0, stream>>>(Y2, scale2, shift2, out);
}